// ResidualGCNLayer_2018634629420
// MI455X (gfx1250) — compile-verified
//
#include <hip/hip_runtime.h>
#include <math.h>

#define IN_C  128
#define HID_C 256
#define OUT_C 256
#define LN_EPS 1e-5f

typedef float v2f __attribute__((ext_vector_type(2)));
typedef float v8f __attribute__((ext_vector_type(8)));

// ---------------------------------------------------------------------------
// degree / dinv helpers
// ---------------------------------------------------------------------------
__global__ void zero_f32_kernel(float* __restrict__ p, int n) {
    int i = blockIdx.x * 256 + threadIdx.x;
    if (i < n) p[i] = 0.0f;
}

__global__ void deg_count_kernel(const long long* __restrict__ col,
                                 float* __restrict__ deg, int E) {
    int e = blockIdx.x * 256 + threadIdx.x;
    if (e < E) atomicAdd(&deg[(int)col[e]], 1.0f);
}

__global__ void deg_to_dinv_kernel(float* __restrict__ d, int n) {
    int i = blockIdx.x * 256 + threadIdx.x;
    if (i < n) d[i] = rsqrtf(d[i] + 1.0f);   // +1 self-loop
}

// ---------------------------------------------------------------------------
// Y[N,256] = X[N,K] @ W[K,256]   (fp32 WMMA 16x16x4)
// grid.x = N/16 row tiles, block = 128 threads = 4 waves.
// Wave w computes the 16x64 strip at columns [w*64, w*64+64).
// A fragment (16x4 f32): lane m=lane&15, kh=lane>>4 holds A[m][kh*2+{0,1}].
// B fragment (4x16 f32): lane holds B[kh*2+{0,1}][m].
// C/D (16x16 f32, 8 VGPRs): vgpr r <-> (M = r + kh*8, N = m).
// ---------------------------------------------------------------------------
__global__ __launch_bounds__(128) void gemm_wmma_f32_kernel(
        const float* __restrict__ X, const float* __restrict__ W,
        float* __restrict__ Y, int K) {
    const int lane = threadIdx.x & 31;
    const int wave = threadIdx.x >> 5;
    const int m    = lane & 15;
    const int kh   = lane >> 4;              // selects K pair {0,1} vs {2,3}
    const int colBase = wave * 64;

    const float* xrow = X + ((size_t)blockIdx.x * 16 + m) * (size_t)K;

    v8f acc0 = {}, acc1 = {}, acc2 = {}, acc3 = {};

    for (int k0 = 0; k0 < K; k0 += 4) {
        const int k = k0 + kh * 2;
        v2f a = *(const v2f*)(xrow + k);     // contiguous K pair -> b64 load

        const float* w0 = W + (size_t)k * 256 + colBase + m;  // row k
        const float* w1 = w0 + 256;                            // row k+1
        v2f b0, b1, b2, b3;
        b0.x = w0[0];  b0.y = w1[0];
        b1.x = w0[16]; b1.y = w1[16];
        b2.x = w0[32]; b2.y = w1[32];
        b3.x = w0[48]; b3.y = w1[48];

        acc0 = __builtin_amdgcn_wmma_f32_16x16x4_f32(false, a, false, b0,
                                                     (short)0, acc0, false, false);
        acc1 = __builtin_amdgcn_wmma_f32_16x16x4_f32(false, a, false, b1,
                                                     (short)0, acc1, false, false);
        acc2 = __builtin_amdgcn_wmma_f32_16x16x4_f32(false, a, false, b2,
                                                     (short)0, acc2, false, false);
        acc3 = __builtin_amdgcn_wmma_f32_16x16x4_f32(false, a, false, b3,
                                                     (short)0, acc3, false, false);
    }

    // Store: M = r + kh*8, N = m (per 16x16 f32 C/D layout)
    float* yb = Y + ((size_t)blockIdx.x * 16 + kh * 8) * 256 + colBase + m;
#pragma unroll
    for (int r = 0; r < 8; ++r) {
        yb[(size_t)r * 256 +  0] = acc0[r];
        yb[(size_t)r * 256 + 16] = acc1[r];
        yb[(size_t)r * 256 + 32] = acc2[r];
        yb[(size_t)r * 256 + 48] = acc3[r];
    }
}

// ---------------------------------------------------------------------------
// agg[i][c] = h[i][c] * dinv[i]^2   (self-loop term, float4 granularity)
// ---------------------------------------------------------------------------
__global__ void selfloop_init_kernel(const float* __restrict__ h,
                                     const float* __restrict__ dinv,
                                     float* __restrict__ agg, int nvec4) {
    int i = blockIdx.x * 256 + threadIdx.x;   // index of a float4 (64 per row)
    if (i >= nvec4) return;
    int node = i >> 6;
    float w = dinv[node]; w *= w;
    float4 v = ((const float4*)h)[i];
    float4 o;
    o.x = v.x * w; o.y = v.y * w; o.z = v.z * w; o.w = v.w * w;
    ((float4*)agg)[i] = o;
}

// ---------------------------------------------------------------------------
// Edge scatter: agg[col] += h[row] * dinv[row]*dinv[col]
// 64 threads per edge, 4 channels (float4) each.
// ---------------------------------------------------------------------------
__global__ __launch_bounds__(256) void scatter_edges_kernel(
        const long long* __restrict__ rowi, const long long* __restrict__ coli,
        const float* __restrict__ dinv, const float* __restrict__ h,
        float* __restrict__ agg, int E) {
    long long t = (long long)blockIdx.x * 256 + threadIdx.x;
    int e = (int)(t >> 6);
    if (e >= E) return;
    int sub = (int)(t & 63);
    int r = (int)rowi[e];
    int c = (int)coli[e];
    float w = dinv[r] * dinv[c];
    float4 v = ((const float4*)(h + (size_t)r * 256))[sub];
    float* dst = agg + (size_t)c * 256 + sub * 4;
    atomicAdd(dst + 0, v.x * w);
    atomicAdd(dst + 1, v.y * w);
    atomicAdd(dst + 2, v.z * w);
    atomicAdd(dst + 3, v.w * w);
}

// ---------------------------------------------------------------------------
// out = relu( LN(agg + bias) * g + bt  [+ resid + resid_bias] )
// One wave per 256-wide row; 8 channels per lane; wave32 xor-shuffle reduce.
// ---------------------------------------------------------------------------
__global__ __launch_bounds__(256) void ln_relu_kernel(
        const float* __restrict__ agg, const float* __restrict__ bias,
        const float* __restrict__ g, const float* __restrict__ bt,
        const float* __restrict__ resid, const float* __restrict__ resid_bias,
        float* __restrict__ out, int N) {
    int lane = threadIdx.x & 31;
    int wave = threadIdx.x >> 5;
    int row  = blockIdx.x * 8 + wave;
    if (row >= N) return;

    const float* rp = agg + (size_t)row * 256;
    float4 x0 = ((const float4*)rp)[lane];
    float4 x1 = ((const float4*)rp)[lane + 32];
    float4 bb0 = ((const float4*)bias)[lane];
    float4 bb1 = ((const float4*)bias)[lane + 32];
    x0.x += bb0.x; x0.y += bb0.y; x0.z += bb0.z; x0.w += bb0.w;
    x1.x += bb1.x; x1.y += bb1.y; x1.z += bb1.z; x1.w += bb1.w;

    float s  = x0.x + x0.y + x0.z + x0.w + x1.x + x1.y + x1.z + x1.w;
    float sq = x0.x*x0.x + x0.y*x0.y + x0.z*x0.z + x0.w*x0.w
             + x1.x*x1.x + x1.y*x1.y + x1.z*x1.z + x1.w*x1.w;
#pragma unroll
    for (int off = 16; off >= 1; off >>= 1) {
        s  += __shfl_xor(s,  off, 32);
        sq += __shfl_xor(sq, off, 32);
    }
    float mu   = s * (1.0f / 256.0f);
    float var  = sq * (1.0f / 256.0f) - mu * mu;
    float rstd = rsqrtf(var + LN_EPS);

    float4 g0 = ((const float4*)g)[lane],  g1 = ((const float4*)g)[lane + 32];
    float4 t0 = ((const float4*)bt)[lane], t1 = ((const float4*)bt)[lane + 32];

    float4 y0, y1;
    y0.x = (x0.x - mu) * rstd * g0.x + t0.x;
    y0.y = (x0.y - mu) * rstd * g0.y + t0.y;
    y0.z = (x0.z - mu) * rstd * g0.z + t0.z;
    y0.w = (x0.w - mu) * rstd * g0.w + t0.w;
    y1.x = (x1.x - mu) * rstd * g1.x + t1.x;
    y1.y = (x1.y - mu) * rstd * g1.y + t1.y;
    y1.z = (x1.z - mu) * rstd * g1.z + t1.z;
    y1.w = (x1.w - mu) * rstd * g1.w + t1.w;

    if (resid != nullptr) {
        const float* rr = resid + (size_t)row * 256;
        float4 r0 = ((const float4*)rr)[lane];
        float4 r1 = ((const float4*)rr)[lane + 32];
        float4 p0 = ((const float4*)resid_bias)[lane];
        float4 p1 = ((const float4*)resid_bias)[lane + 32];
        y0.x += r0.x + p0.x; y0.y += r0.y + p0.y;
        y0.z += r0.z + p0.z; y0.w += r0.w + p0.w;
        y1.x += r1.x + p1.x; y1.y += r1.y + p1.y;
        y1.z += r1.z + p1.z; y1.w += r1.w + p1.w;
    }

    y0.x = fmaxf(y0.x, 0.0f); y0.y = fmaxf(y0.y, 0.0f);
    y0.z = fmaxf(y0.z, 0.0f); y0.w = fmaxf(y0.w, 0.0f);
    y1.x = fmaxf(y1.x, 0.0f); y1.y = fmaxf(y1.y, 0.0f);
    y1.z = fmaxf(y1.z, 0.0f); y1.w = fmaxf(y1.w, 0.0f);

    float* op = out + (size_t)row * 256;
    ((float4*)op)[lane]      = y0;
    ((float4*)op)[lane + 32] = y1;
}

// ---------------------------------------------------------------------------
extern "C" void kernel_launch(void* const* d_in, const int* in_sizes, int n_in,
                              void* d_out, int out_size, void* d_ws, size_t ws_size,
                              hipStream_t stream) {
    const float*     x    = (const float*)d_in[0];
    const long long* ei   = (const long long*)d_in[1];   // int64 [2, E]
    const float*     W1   = (const float*)d_in[2];
    const float*     b1   = (const float*)d_in[3];
    const float*     W2   = (const float*)d_in[4];
    const float*     b2   = (const float*)d_in[5];
    const float*     l1g  = (const float*)d_in[6];
    const float*     l1b  = (const float*)d_in[7];
    const float*     l2g  = (const float*)d_in[8];
    const float*     l2b  = (const float*)d_in[9];
    const float*     pW   = (const float*)d_in[10];
    const float*     pb   = (const float*)d_in[11];

    const int N = in_sizes[0] / IN_C;
    const int E = in_sizes[1] / 2;
    const long long* rowi = ei;
    const long long* coli = ei + E;

    float* ws   = (float*)d_ws;
    size_t npad = (((size_t)N + 1023) / 1024) * 1024;
    float* dinv = ws;
    float* bufA = ws + npad;                       // [N,256]
    float* bufB = bufA + (size_t)N * 256;          // [N,256]
    float* bufD = bufB + (size_t)N * 256;          // [N,256]
    float* out  = (float*)d_out;

    const int nvec4 = N * 64;                      // float4s per [N,256]

    // degrees -> dinv
    zero_f32_kernel<<<(N + 255) / 256, 256, 0, stream>>>(dinv, N);
    deg_count_kernel<<<(E + 255) / 256, 256, 0, stream>>>(coli, dinv, E);
    deg_to_dinv_kernel<<<(N + 255) / 256, 256, 0, stream>>>(dinv, N);

    // layer 1: h1 = x @ W1 -> bufA ; agg1 -> bufB
    gemm_wmma_f32_kernel<<<N / 16, 128, 0, stream>>>(x, W1, bufA, IN_C);
    selfloop_init_kernel<<<(nvec4 + 255) / 256, 256, 0, stream>>>(bufA, dinv, bufB, nvec4);
    scatter_edges_kernel<<<(int)(((long long)E * 64 + 255) / 256), 256, 0, stream>>>(
        rowi, coli, dinv, bufA, bufB, E);
    // bufA = relu(LN(agg1 + b1))
    ln_relu_kernel<<<(N + 7) / 8, 256, 0, stream>>>(bufB, b1, l1g, l1b,
                                                    nullptr, nullptr, bufA, N);

    // layer 2: h2 = bufA @ W2 -> bufB ; agg2 -> bufD
    gemm_wmma_f32_kernel<<<N / 16, 128, 0, stream>>>(bufA, W2, bufB, HID_C);
    selfloop_init_kernel<<<(nvec4 + 255) / 256, 256, 0, stream>>>(bufB, dinv, bufD, nvec4);
    scatter_edges_kernel<<<(int)(((long long)E * 64 + 255) / 256), 256, 0, stream>>>(
        rowi, coli, dinv, bufB, bufD, E);

    // identity = x @ proj_W -> d_out (proj_b folded into final epilogue)
    gemm_wmma_f32_kernel<<<N / 16, 128, 0, stream>>>(x, pW, out, IN_C);

    // out = relu(LN(agg2 + b2)*g + bt + identity + proj_b)
    ln_relu_kernel<<<(N + 7) / 8, 256, 0, stream>>>(bufD, b2, l2g, l2b,
                                                    out, pb, out, N);
}